// TopkSeqLatentMasker_54984171324132
// MI455X (gfx1250) — compile-verified
//
#include <hip/hip_runtime.h>
#include <stdint.h>

#define SDIM 4096
#define KTOP 64
#define NT   256
#define CAP  1024   // candidate buffer; expected occupancy ~70 of 1024

typedef float __attribute__((ext_vector_type(4))) fvec4;   // native vector for NT stores

// ---------------------------------------------------------------------------
// Kernel 1: fill the whole [B,1,S,S] mask with 1.0f (pure bandwidth, 128-bit
// non-temporal stores -> don't thrash the 192MB L2 with a 256MB stream).
// ---------------------------------------------------------------------------
__global__ void init_mask_kernel(float* __restrict__ out, size_t n4) {
    size_t i      = (size_t)blockIdx.x * blockDim.x + threadIdx.x;
    size_t stride = (size_t)gridDim.x * blockDim.x;
    fvec4 one = {1.f, 1.f, 1.f, 1.f};
    fvec4* o = reinterpret_cast<fvec4*>(out);
    for (; i < n4; i += stride) __builtin_nontemporal_store(one, &o[i]);
}

// Order-preserving fp32 -> u32 map (larger float => larger key).
__device__ __forceinline__ uint32_t orderable(uint32_t u) {
    return (u & 0x80000000u) ? ~u : (u | 0x80000000u);
}

// ---------------------------------------------------------------------------
// Kernel 2: one workgroup per row. Async-copy row to LDS (non-temporal),
// 2-level radix select -> 16-bit threshold, collect candidates, bitonic sort,
// scatter 0.0f with the (k+1)-triangle writes suppressed.
// ---------------------------------------------------------------------------
__global__ void __launch_bounds__(NT)
topk_scatter_kernel(const float* __restrict__ scores, float* __restrict__ out) {
    __shared__ __align__(16) uint32_t key[SDIM];        // 16 KB (row as keys)
    __shared__ uint32_t hist[256];                      // 1 KB
    __shared__ uint32_t sca[256];                       // 1 KB suffix-scan
    __shared__ unsigned long long cand[CAP];            // 8 KB
    __shared__ uint32_t cnt;
    __shared__ uint32_t sh_t1, sh_above, sh_key16;

    const int tid = threadIdx.x;
    const int row = blockIdx.x;                 // row = b*S + s
    const int b   = row >> 12;                  // /4096
    const size_t base = (size_t)row * SDIM;

    // ---- stage row into LDS via async global->LDS (16 KB = 1024 x b128) ----
    {
        const uint64_t gbase = (uint64_t)(const void*)(scores + base);
        const uint32_t lbase = (uint32_t)(size_t)(&key[0]);   // LDS byte offset
        #pragma unroll
        for (int i = 0; i < 4; ++i) {
            int c = tid + NT * i;                              // chunk 0..1023
            uint64_t ga = gbase + (uint64_t)c * 16u;
            uint32_t la = lbase + (uint32_t)c * 16u;
            asm volatile("global_load_async_to_lds_b128 %0, %1, off th:TH_LOAD_NT"
                         :: "v"(la), "v"(ga) : "memory");
        }
        asm volatile("s_wait_asynccnt 0" ::: "memory");
    }
    __syncthreads();

    // ---- transform in place to orderable keys ----
    for (int i = tid; i < SDIM; i += NT) key[i] = orderable(key[i]);
    hist[tid] = 0;
    __syncthreads();

    // ---- pass 1: histogram of top 8 key bits ----
    for (int i = tid; i < SDIM; i += NT) atomicAdd(&hist[key[i] >> 24], 1u);
    __syncthreads();
    // suffix sum: sca[d] = #elements with digit >= d
    sca[tid] = hist[tid];
    __syncthreads();
    for (int off = 1; off < 256; off <<= 1) {
        uint32_t add = (tid + off < 256) ? sca[tid + off] : 0u;
        __syncthreads();
        sca[tid] += add;
        __syncthreads();
    }
    // T1 = max d with suffix[d] >= K  (bucket of the 64th largest)
    if (sca[tid] >= KTOP && (tid == 255 || sca[tid + 1] < KTOP)) {
        sh_t1    = (uint32_t)tid;
        sh_above = (tid == 255) ? 0u : sca[tid + 1];
    }
    __syncthreads();
    const uint32_t T1 = sh_t1;
    const uint32_t k2 = KTOP - sh_above;      // rank needed inside bucket T1
    hist[tid] = 0;
    __syncthreads();

    // ---- pass 2: histogram of bits 23:16 restricted to bucket T1 ----
    for (int i = tid; i < SDIM; i += NT) {
        uint32_t kk = key[i];
        if ((kk >> 24) == T1) atomicAdd(&hist[(kk >> 16) & 0xFFu], 1u);
    }
    __syncthreads();
    sca[tid] = hist[tid];
    __syncthreads();
    for (int off = 1; off < 256; off <<= 1) {
        uint32_t add = (tid + off < 256) ? sca[tid + off] : 0u;
        __syncthreads();
        sca[tid] += add;
        __syncthreads();
    }
    if (sca[tid] >= k2 && (tid == 255 || sca[tid + 1] < k2)) {
        sh_key16 = (T1 << 8) | (uint32_t)tid;   // 16-bit key of 64th element
    }
    if (tid == 0) cnt = 0;
    __syncthreads();
    const uint32_t key16 = sh_key16;

    // ---- pass 3: collect candidates (top16 >= threshold) ----
    // combined key: value desc, then index asc on ties (JAX top_k semantics)
    for (int i = tid; i < SDIM; i += NT) {
        uint32_t kk = key[i];
        if ((kk >> 16) >= key16) {
            uint32_t p = atomicAdd(&cnt, 1u);
            if (p < CAP)
                cand[p] = ((unsigned long long)kk << 32) | (uint32_t)(~(uint32_t)i);
        }
    }
    __syncthreads();
    uint32_t n = cnt; if (n > CAP) n = CAP;     // n >= 64 by construction
    uint32_t P = 64; while (P < n) P <<= 1;     // pad size (<= CAP)
    for (uint32_t i = n + tid; i < P; i += NT) cand[i] = 0ull;

    // ---- bitonic sort, descending ----
    for (uint32_t size = 2; size <= P; size <<= 1) {
        for (uint32_t stride = size >> 1; stride > 0; stride >>= 1) {
            __syncthreads();
            for (uint32_t i = tid; i < P; i += NT) {
                uint32_t j2 = i ^ stride;
                if (j2 > i) {
                    bool up = ((i & size) == 0);      // 'up' block = descending
                    unsigned long long a = cand[i], c = cand[j2];
                    if (up ? (a < c) : (a > c)) { cand[i] = c; cand[j2] = a; }
                }
            }
        }
    }
    __syncthreads();

    // ---- scatter: mask[b, r, j] = 0, suppressed inside the (k+1) triangle ----
    if (tid < KTOP) {
        unsigned long long e = cand[tid];
        uint32_t r = ~((uint32_t)e);                 // recover element index
        uint32_t j = (uint32_t)tid;                  // rank -> column
        if (r < SDIM && !(r < (KTOP + 1) && j > r)) {
            out[(size_t)b * SDIM * SDIM + (size_t)r * SDIM + j] = 0.0f;
        }
    }
}

// ---------------------------------------------------------------------------
extern "C" void kernel_launch(void* const* d_in, const int* in_sizes, int n_in,
                              void* d_out, int out_size, void* d_ws, size_t ws_size,
                              hipStream_t stream) {
    const float* scores = (const float*)d_in[0];   // [B,S,S] fp32
    float* out = (float*)d_out;                    // [B,1,S,S] as float 0/1

    size_t n4 = (size_t)out_size / 4;              // float4 count
    init_mask_kernel<<<4096, NT, 0, stream>>>(out, n4);

    int rows = in_sizes[0] / SDIM;                 // B*S = 16384
    topk_scatter_kernel<<<rows, NT, 0, stream>>>(scores, out);
}